// PatchNetVLAD_11776800325970
// MI455X (gfx1250) — compile-verified
//
#include <hip/hip_runtime.h>
#include <math.h>

typedef float v8f __attribute__((ext_vector_type(8)));
typedef float v2f __attribute__((ext_vector_type(2)));

#define N_   4
#define C_   128
#define H_   40
#define W_   30
#define K_   64
#define HW_  (H_ * W_)        // 1200
#define P_   5
#define HO_  (H_ + 1 - P_)    // 36
#define WO_  (W_ + 1 - P_)    // 26
#define MT_  (HW_ / 16)       // 75 m-tiles per image

// ---------------------------------------------------------------------------
// Kernel 1: descriptor-wise L2 normalization. xn = x / max(||x||_2, 1e-12).
// One thread per (n,h,w); loads are coalesced (consecutive lanes -> consecutive
// spatial positions; channel stride HW_ between loop iterations).
// ---------------------------------------------------------------------------
__global__ __launch_bounds__(256) void k_normalize(const float* __restrict__ x,
                                                   float* __restrict__ xn) {
  int p = blockIdx.x * blockDim.x + threadIdx.x;
  if (p >= N_ * HW_) return;
  int n = p / HW_;
  int t = p - n * HW_;
  const float* xb = x + (size_t)n * C_ * HW_ + t;
  float s = 0.0f;
#pragma unroll 4
  for (int c = 0; c < C_; ++c) {
    float v = xb[(size_t)c * HW_];
    s += v * v;
  }
  float scale = 1.0f / fmaxf(sqrtf(s), 1e-12f);
  float* ob = xn + (size_t)n * C_ * HW_ + t;
#pragma unroll 4
  for (int c = 0; c < C_; ++c) ob[(size_t)c * HW_] = xb[(size_t)c * HW_] * scale;
}

// ---------------------------------------------------------------------------
// Kernel 2: logits = xn^T * W (per image: M=1200, N=K=64, inner=C=128) using
// V_WMMA_F32_16X16X4_F32 (full fp32 precision; compute is free, stay exact),
// then softmax over K=64 -> sa[N,K,H,W].
// One wave (32 threads) per 16-row m-tile. A-matrix layout per CDNA5 ISA:
// lanes 0-15 hold inner-K {0,1} in VGPR{0,1}, lanes 16-31 hold inner-K {2,3}.
// ---------------------------------------------------------------------------
__global__ __launch_bounds__(32) void k_logits_softmax(const float* __restrict__ xn,
                                                       const float* __restrict__ w,
                                                       float* __restrict__ sa) {
  int bid = blockIdx.x;
  int n = bid / MT_;
  int mt = bid - n * MT_;
  int pos0 = mt * 16;
  int lane = threadIdx.x;
  int m = lane & 15;
  int half = lane >> 4;

  v8f acc0 = {}, acc1 = {}, acc2 = {}, acc3 = {};
  const float* xb = xn + (size_t)n * C_ * HW_ + pos0 + m;

#pragma unroll 4
  for (int cc = 0; cc < C_; cc += 4) {
    int ca = cc + half * 2;  // this lane's inner-K base (0/2 offset by half)
    v2f a;
    a.x = xb[(size_t)ca * HW_];
    a.y = xb[(size_t)(ca + 1) * HW_];

    v2f b0, b1, b2, b3;
    b0.x = w[(0 * 16 + m) * C_ + ca]; b0.y = w[(0 * 16 + m) * C_ + ca + 1];
    b1.x = w[(1 * 16 + m) * C_ + ca]; b1.y = w[(1 * 16 + m) * C_ + ca + 1];
    b2.x = w[(2 * 16 + m) * C_ + ca]; b2.y = w[(2 * 16 + m) * C_ + ca + 1];
    b3.x = w[(3 * 16 + m) * C_ + ca]; b3.y = w[(3 * 16 + m) * C_ + ca + 1];

    acc0 = __builtin_amdgcn_wmma_f32_16x16x4_f32(false, a, false, b0, (short)0, acc0, false, false);
    acc1 = __builtin_amdgcn_wmma_f32_16x16x4_f32(false, a, false, b1, (short)0, acc1, false, false);
    acc2 = __builtin_amdgcn_wmma_f32_16x16x4_f32(false, a, false, b2, (short)0, acc2, false, false);
    acc3 = __builtin_amdgcn_wmma_f32_16x16x4_f32(false, a, false, b3, (short)0, acc3, false, false);
  }

  // Spill 16x64 logits to LDS (C/D layout: VGPR v, lanes<16 -> M=v, lanes>=16 -> M=v+8)
  __shared__ float lg[16 * K_];
#pragma unroll
  for (int v = 0; v < 8; ++v) {
    int row = v + 8 * half;
    lg[row * K_ + 0 * 16 + m] = acc0[v];
    lg[row * K_ + 1 * 16 + m] = acc1[v];
    lg[row * K_ + 2 * 16 + m] = acc2[v];
    lg[row * K_ + 3 * 16 + m] = acc3[v];
  }
  __syncthreads();

  if (lane < 16) {
    float* row = lg + lane * K_;
    float mx = -3.402823466e+38f;
    for (int k = 0; k < K_; ++k) mx = fmaxf(mx, row[k]);
    float s = 0.0f;
    for (int k = 0; k < K_; ++k) {
      float e = __expf(row[k] - mx);
      row[k] = e;
      s += e;
    }
    float inv = 1.0f / s;
    for (int k = 0; k < K_; ++k)
      sa[((size_t)(n * K_ + k)) * HW_ + pos0 + lane] = row[k] * inv;
  }
}

// ---------------------------------------------------------------------------
// Kernel 3: out[n,k,c,i,j] = (boxsum(sa_k * xn_c) - cent[k,c]*boxsum(sa_k))/25
// One workgroup per (n, i). Stage 5 sa-rows for all K (38.4KB) + S2 (6.7KB) +
// a 16-channel xn chunk (9.6KB) in LDS; sliding-window over j with float2
// stores (8B-aligned: 936*4 and i*104 are both multiples of 8).
// ---------------------------------------------------------------------------
__global__ __launch_bounds__(256) void k_patch(const float* __restrict__ xn,
                                               const float* __restrict__ sa,
                                               const float* __restrict__ cent,
                                               float* __restrict__ out) {
  __shared__ float s_sa[K_ * P_ * W_];   // 64*150 floats = 38400 B
  __shared__ float s_xn[16 * P_ * W_];   // 16*150 floats =  9600 B
  __shared__ float s_S2[K_ * WO_];       // 64*26  floats =  6656 B

  int bid = blockIdx.x;
  int n = bid / HO_;
  int i = bid - n * HO_;
  int tid = threadIdx.x;

  // Load 5 rows of sa for all 64 clusters.
  for (int idx = tid; idx < K_ * P_ * W_; idx += 256) {
    int k = idx / (P_ * W_);
    int r = idx - k * (P_ * W_);
    int dh = r / W_;
    int ww = r - dh * W_;
    s_sa[idx] = sa[((size_t)(n * K_ + k)) * HW_ + (i + dh) * W_ + ww];
  }
  __syncthreads();

  // S2[k][j] = 5x5 box sum of sa (depends only on k).
  if (tid < K_) {
    int k = tid;
    float cs[W_];
#pragma unroll
    for (int ww = 0; ww < W_; ++ww) {
      float s = 0.0f;
#pragma unroll
      for (int dh = 0; dh < P_; ++dh) s += s_sa[k * (P_ * W_) + dh * W_ + ww];
      cs[ww] = s;
    }
    float run = cs[0] + cs[1] + cs[2] + cs[3] + cs[4];
#pragma unroll
    for (int j = 0; j < WO_; ++j) {
      s_S2[k * WO_ + j] = run;
      if (j < WO_ - 1) run += cs[j + P_] - cs[j];
    }
  }
  __syncthreads();

  for (int ch = 0; ch < C_ / 16; ++ch) {
    int c0 = ch * 16;
    // Load 5 rows x 16 channels of xn.
    for (int idx = tid; idx < 16 * P_ * W_; idx += 256) {
      int c = idx / (P_ * W_);
      int r = idx - c * (P_ * W_);
      int dh = r / W_;
      int ww = r - dh * W_;
      s_xn[idx] = xn[((size_t)(n * C_ + c0 + c)) * HW_ + (i + dh) * W_ + ww];
    }
    // Prefetch next chunk's rows into cache (gfx1250 global_prefetch_b8).
    if (ch + 1 < C_ / 16) {
      int pc = 16 + (tid >> 4);              // next chunk channel
      int po = (tid & 15) * 10;              // spread across the 600B row span
      __builtin_prefetch(&xn[((size_t)(n * C_ + c0 + pc)) * HW_ + i * W_ + po], 0, 3);
    }
    __syncthreads();

#pragma unroll 1
    for (int it = 0; it < 4; ++it) {
      int p = tid + it * 256;      // 0..1023 -> (k, c) pair
      int k = p >> 4;
      int c = p & 15;
      float ckc = cent[k * C_ + c0 + c];

      // Vertical 5-row weighted sums per column.
      float col[W_];
#pragma unroll
      for (int ww = 0; ww < W_; ++ww) {
        float s = 0.0f;
#pragma unroll
        for (int dh = 0; dh < P_; ++dh)
          s = fmaf(s_sa[k * (P_ * W_) + dh * W_ + ww],
                   s_xn[c * (P_ * W_) + dh * W_ + ww], s);
        col[ww] = s;
      }

      float run = col[0] + col[1] + col[2] + col[3] + col[4];
      float* op = out + ((size_t)((n * K_ + k) * C_ + c0 + c)) * (HO_ * WO_) + (size_t)i * WO_;
      const float inv25 = 1.0f / (P_ * P_);
#pragma unroll
      for (int j2 = 0; j2 < WO_ / 2; ++j2) {
        int j = 2 * j2;
        float o0 = (run - ckc * s_S2[k * WO_ + j]) * inv25;
        run += col[j + P_] - col[j];
        float o1 = (run - ckc * s_S2[k * WO_ + j + 1]) * inv25;
        if (j + 1 < WO_ - 1) run += col[j + 1 + P_] - col[j + 1];
        *(float2*)(op + j) = make_float2(o0, o1);
      }
    }
    __syncthreads();
  }
}

// ---------------------------------------------------------------------------
extern "C" void kernel_launch(void* const* d_in, const int* in_sizes, int n_in,
                              void* d_out, int out_size, void* d_ws, size_t ws_size,
                              hipStream_t stream) {
  const float* x    = (const float*)d_in[0];  // [N,C,H,W]
  const float* w    = (const float*)d_in[1];  // [K,C]
  const float* cent = (const float*)d_in[2];  // [K,C]
  // d_in[3] = patch_size(5), d_in[4] = stride(1): fixed scalars, hardcoded.
  float* out = (float*)d_out;

  float* xn = (float*)d_ws;                    // N*C*HW = 614400 floats
  float* sa = xn + (size_t)N_ * C_ * HW_;      // N*K*HW = 307200 floats

  k_normalize<<<(N_ * HW_ + 255) / 256, 256, 0, stream>>>(x, xn);
  k_logits_softmax<<<N_ * MT_, 32, 0, stream>>>(xn, w, sa);
  k_patch<<<N_ * HO_, 256, 0, stream>>>(xn, sa, cent, out);
}